// SmoothAP_68745246540428
// MI455X (gfx1250) — compile-verified
//
#include <hip/hip_runtime.h>
#include <hip/hip_bf16.h>

#define NN 512   // anchors
#define DD 512   // feature dim

typedef __attribute__((ext_vector_type(2))) float v2f;
typedef __attribute__((ext_vector_type(8))) float v8f;

// ---------------------------------------------------------------------------
// Kernel 1: sim = preds * preds^T via V_WMMA_F32_16X16X4_F32.
// One wave per 16x16 output tile; 32x32 = 1024 tiles; 4 waves/block.
//
// f32 WMMA VGPR layout (ISA 7.12.2):
//   A (16x4, MxK):  lanes 0-15 hold K={0,1} in v[0],v[1]; lanes 16-31 K={2,3}
//   B (4x16, KxN):  rows striped across lanes; same half-split on K
//   For C = P*P^T (B[k,j] = P[j,k]) the A- and B-fragments are the SAME
//   gather pattern applied to the row tile / column tile respectively.
//   C/D (16x16):    v[r] holds row (r + 8*half), col = lane&15
// ---------------------------------------------------------------------------
__global__ __launch_bounds__(128) void wmma_gram(const float* __restrict__ preds,
                                                 float* __restrict__ sim) {
  const int wave = threadIdx.x >> 5;
  const int lane = threadIdx.x & 31;
  const int tile = blockIdx.x * 4 + wave;   // 0..1023
  const int ti = tile >> 5;                 // tile row
  const int tj = tile & 31;                 // tile col
  const int i0 = ti * 16, j0 = tj * 16;
  const int half = lane >> 4;               // K half-select per ISA layout
  const int l16  = lane & 15;

  const float* arow = preds + (size_t)(i0 + l16) * DD;  // A row for this lane
  const float* brow = preds + (size_t)(j0 + l16) * DD;  // B "column" = P row j

  v8f c = {};
  for (int k = 0; k < DD; k += 4) {
    const int kk = k + 2 * half;
    v2f a = *(const v2f*)(arow + kk);   // global_load_b64
    v2f b = *(const v2f*)(brow + kk);   // global_load_b64
    // 8 args: (neg_a, A, neg_b, B, c_mod, C, reuse_a, reuse_b)
    c = __builtin_amdgcn_wmma_f32_16x16x4_f32(false, a, false, b, (short)0, c,
                                              false, false);
  }

  float* out = sim + (size_t)(i0 + 8 * half) * NN + (j0 + l16);
#pragma unroll
  for (int r = 0; r < 8; ++r) out[(size_t)r * NN] = c[r];
}

// ---------------------------------------------------------------------------
// Kernel 2: per-anchor SmoothAP reduction, exploiting numerator sparsity.
// One block (8 waves) per anchor i. Only j == i and positive j are evaluated.
// For each such j:  A = sum_{k!=j} sig(100*(s_ik - s_ij))
//                   P = sum_{k!=j, pos(i,k)} sig(...)
//   term(j==i)  = 1/(1+A)
//   term(j pos) = (1+P)/(1+A)
// per_anchor[i] = (sum_j term) / (m+1), zero if no positives (n_pos<=1).
// sigmoid via v_exp_f32/v_rcp_f32; 1/temp and log2(e) folded into one scale.
// ---------------------------------------------------------------------------
__global__ __launch_bounds__(256) void smoothap_rows(const float* __restrict__ sim,
                                                     const int* __restrict__ labels,
                                                     float* __restrict__ per_anchor) {
  __shared__ float srow[NN];
  __shared__ float smask[NN];
  __shared__ short poslist[NN];
  __shared__ int   pcount;
  __shared__ float wavesum[8];

  const int i   = blockIdx.x;
  const int tid = threadIdx.x;
  const int li  = labels[i];

  for (int k = tid; k < NN; k += 256) {
    srow[k]  = sim[(size_t)i * NN + k];
    smask[k] = (labels[k] == li && k != i) ? 1.0f : 0.0f;
  }
  __syncthreads();

  // Deterministic positives list (single-thread scan; 512 iters, negligible).
  if (tid == 0) {
    int m = 0;
    for (int k = 0; k < NN; ++k)
      if (smask[k] != 0.0f) poslist[m++] = (short)k;
    pcount = m;
  }
  __syncthreads();
  const int m = pcount;

  const int wave = tid >> 5, lane = tid & 31;
  // clip(x*100, -50, 50) then *log2e  ==  clip(x*100*log2e, +/-50*log2e)
  const float SCALE = 100.0f * 1.4426950408889634f;
  const float CLIP  = 50.0f * 1.4426950408889634f;

  float wsum = 0.0f;
  for (int job = wave; job <= m; job += 8) {     // job 0 -> j=i, else positives
    const int   j  = (job == 0) ? i : (int)poslist[job - 1];
    const float sj = srow[j];
    float A = 0.0f, P = 0.0f;
    for (int k = lane; k < NN; k += 32) {
      float u = (srow[k] - sj) * SCALE;
      u = fminf(fmaxf(u, -CLIP), CLIP);
      float sg = __builtin_amdgcn_rcpf(1.0f + __builtin_amdgcn_exp2f(-u));
      A += sg;
      P += sg * smask[k];
    }
#pragma unroll
    for (int off = 16; off > 0; off >>= 1) {     // wave32 reduce
      A += __shfl_xor(A, off, 32);
      P += __shfl_xor(P, off, 32);
    }
    if (lane == 0) {
      A -= 0.5f;                // remove k==j term (sigmoid(0)=0.5)
      P -= 0.5f * smask[j];     // ditto if j itself is a positive
      wsum += (job == 0) ? 1.0f / (1.0f + A) : (1.0f + P) / (1.0f + A);
    }
  }
  if (lane == 0) wavesum[wave] = wsum;
  __syncthreads();

  if (tid == 0) {
    float s = 0.0f;
    for (int w = 0; w < 8; ++w) s += wavesum[w];
    per_anchor[i] = (m > 0) ? s / (float)(m + 1) : 0.0f;
  }
}

// ---------------------------------------------------------------------------
// Kernel 3: deterministic fixed-order final sum; out = 1 - ap.
// ---------------------------------------------------------------------------
__global__ void smoothap_finalize(const float* __restrict__ per_anchor,
                                  float* __restrict__ out) {
  if (threadIdx.x == 0) {
    float s = 0.0f;
    for (int k = 0; k < NN; ++k) s += per_anchor[k];
    out[0] = 1.0f - s / (float)NN;
  }
}

extern "C" void kernel_launch(void* const* d_in, const int* in_sizes, int n_in,
                              void* d_out, int out_size, void* d_ws, size_t ws_size,
                              hipStream_t stream) {
  const float* preds  = (const float*)d_in[0];
  const int*   labels = (const int*)d_in[1];

  float* sim        = (float*)d_ws;        // 512*512 f32 = 1 MB
  float* per_anchor = sim + NN * NN;       // + 512 f32
  float* out        = (float*)d_out;

  wmma_gram<<<256, 128, 0, stream>>>(preds, sim);
  smoothap_rows<<<NN, 256, 0, stream>>>(sim, labels, per_anchor);
  smoothap_finalize<<<1, 32, 0, stream>>>(per_anchor, out);
}